// TTTAdapter_85057532330591
// MI455X (gfx1250) — compile-verified
//
#include <hip/hip_runtime.h>
#include <math.h>

typedef __attribute__((ext_vector_type(2))) float v2f;
typedef __attribute__((ext_vector_type(8))) float v8f;

// ---------------------------------------------------------------------------
// Generic fp32 WMMA GEMM: C[MxN] = A[MxK] @ B[KxN]  (+bias) (+silu)
// One 16x16 output tile per wave, K advanced 4 per V_WMMA_F32_16X16X4_F32.
// EPI: 0 = none, 1 = +bias, 2 = +bias then silu
// Requires M%16==0 (grid.y*4 waves), N%16==0, K%4==0; EXEC all-ones (no
// divergence) as WMMA demands.
// ---------------------------------------------------------------------------
template <int EPI>
__global__ __launch_bounds__(128) void wmma_gemm_f32(
    const float* __restrict__ A, const float* __restrict__ B,
    const float* __restrict__ bias, float* __restrict__ C,
    int M, int N, int K)
{
    const int lane = threadIdx.x & 31;
    const int wave = threadIdx.x >> 5;
    const int n0   = blockIdx.x * 16;
    const int m0   = (blockIdx.y * 4 + wave) * 16;
    const int half = lane >> 4;   // 0: K pair {0,1}, 1: K pair {2,3}
    const int l    = lane & 15;   // M index for A frag, N index for B frag

    v8f acc = {};
    const float* Arow = A + (size_t)(m0 + l) * K;
    for (int k0 = 0; k0 < K; k0 += 4) {
        const int ka = k0 + half * 2;
        v2f av, bv;
        av.x = Arow[ka];
        av.y = Arow[ka + 1];
        bv.x = B[(size_t)ka       * N + n0 + l];
        bv.y = B[(size_t)(ka + 1) * N + n0 + l];
        acc = __builtin_amdgcn_wmma_f32_16x16x4_f32(
            /*neg_a=*/false, av, /*neg_b=*/false, bv,
            /*c_mod=*/(short)0, acc, /*reuse_a=*/false, /*reuse_b=*/false);
    }

    const int col = n0 + l;
    const float bs = (EPI >= 1) ? bias[col] : 0.0f;
#pragma unroll
    for (int i = 0; i < 8; ++i) {
        const int row = m0 + i + half * 8;      // VGPR i: M=i (lanes 0-15), M=i+8 (16-31)
        float v = acc[i] + bs;
        if (EPI == 2) v = v / (1.0f + expf(-v));   // silu
        C[(size_t)row * N + col] = v;
    }
}

// ---------------------------------------------------------------------------
// eta = sigmoid(h @ lr_w + lr_b) / hd       (M x H), H=3 so plain VALU kernel
// ---------------------------------------------------------------------------
__global__ void eta_kernel(const float* __restrict__ Hh,
                           const float* __restrict__ lr_w,
                           const float* __restrict__ lr_b,
                           float* __restrict__ ETA, int M, int A, int H)
{
    const int i = blockIdx.x * blockDim.x + threadIdx.x;
    if (i >= M * H) return;
    const int r = i / H, h = i % H;
    float acc = lr_b[h];
    const float* row = Hh + (size_t)r * A;
    for (int a = 0; a < A; ++a) acc += row[a] * lr_w[a * H + h];
    ETA[i] = (1.0f / (1.0f + expf(-acc))) * (1.0f / 64.0f);   // hd = 64
}

// ---------------------------------------------------------------------------
// Row-wise LayerNorm over A=192: one block of 192 threads (6 waves) per row.
// ---------------------------------------------------------------------------
__global__ __launch_bounds__(192) void ln_rows(const float* __restrict__ X,
                                               const float* __restrict__ g,
                                               const float* __restrict__ bta,
                                               float* __restrict__ Y, int A)
{
    const int r = blockIdx.x, t = threadIdx.x;
    const int wid = t >> 5, lid = t & 31;
    __shared__ float sred[12];

    const float x = X[(size_t)r * A + t];
    float a = x, b2 = x * x;
#pragma unroll
    for (int off = 16; off > 0; off >>= 1) {
        a  += __shfl_xor(a,  off, 32);
        b2 += __shfl_xor(b2, off, 32);
    }
    if (lid == 0) { sred[wid * 2] = a; sred[wid * 2 + 1] = b2; }
    __syncthreads();
    float s1 = 0.f, s2 = 0.f;
#pragma unroll
    for (int w = 0; w < 6; ++w) { s1 += sred[w * 2]; s2 += sred[w * 2 + 1]; }
    const float inv  = 1.0f / (float)A;
    const float mu   = s1 * inv;
    const float var  = s2 * inv - mu * mu;
    const float istd = rsqrtf(var + 1e-6f);
    Y[(size_t)r * A + t] = (x - mu) * istd * g[t] + bta[t];
}

// ---------------------------------------------------------------------------
// Sequential TTT scan. ONE WAVE (32 lanes) per (b,h) chain; lane owns
// columns e and e+32 of the 64x64 fast weight (128 VGPRs of state + cached
// k vector). All LN reductions are pure wave32 shfl_xor chains -> no LDS
// exchanges, and in single-wave workgroups barriers cost nothing.
// Next-step q/k/v/eta are prefetched into registers while the current step
// computes (data lives in L2; this hides the per-step load latency), and
// step s+2 is pulled in with global_prefetch_b8.
// Algebraic refactor: Zq = (q.W_old + b_old) - eta*g*(q.k + 1), so both
// mat-vecs run before the LN-grad dependency.
// ---------------------------------------------------------------------------
__device__ __forceinline__ void wave_red2(float& a, float& b)
{
#pragma unroll
    for (int off = 16; off > 0; off >>= 1) {
        a += __shfl_xor(a, off, 32);
        b += __shfl_xor(b, off, 32);
    }
}

__global__ __launch_bounds__(32) void ttt_scan(
    const float* __restrict__ Q, const float* __restrict__ Kx,
    const float* __restrict__ V, const float* __restrict__ ETA,
    const float* __restrict__ W1m, const float* __restrict__ b1,
    const float* __restrict__ ttt_g, const float* __restrict__ ttt_b,
    float* __restrict__ YS, int S, int H, int A)
{
    const int b    = blockIdx.x / H;
    const int h    = blockIdx.x % H;
    const int lane = threadIdx.x;        // 0..31
    const int e0   = lane, e1 = lane + 32;

    __shared__ float4 skv4[16], sqv4[16];
    float* sk = (float*)skv4;
    float* sq = (float*)sqv4;

    float Wa[64], Wb[64], kreg[64];      // columns e0/e1 of W + cached k
#pragma unroll
    for (int d = 0; d < 64; ++d) {
        Wa[d] = W1m[h * 4096 + d * 64 + e0];
        Wb[d] = W1m[h * 4096 + d * 64 + e1];
    }
    float bc0 = b1[h * 64 + e0], bc1 = b1[h * 64 + e1];
    const float gam0 = ttt_g[h * 64 + e0], gam1 = ttt_g[h * 64 + e1];
    const float bet0 = ttt_b[h * 64 + e0], bet1 = ttt_b[h * 64 + e1];

    // preload step 0
    size_t base0 = (size_t)(b * S) * A + h * 64;
    float kc0 = Kx[base0 + e0], kc1 = Kx[base0 + e1];
    float qc0 = Q[base0 + e0],  qc1 = Q[base0 + e1];
    float vc0 = V[base0 + e0],  vc1 = V[base0 + e1];
    float etac = ETA[(size_t)(b * S) * H + h];

    for (int s = 0; s < S; ++s) {
        const size_t baseo = (size_t)(b * S + s) * A + h * 64;
        sk[e0] = kc0; sk[e1] = kc1;
        sq[e0] = qc0; sq[e1] = qc1;
        __syncthreads();                 // single wave: just a DS fence

        // prefetch step s+1 into registers (hidden behind this step's math)
        const int sn = (s + 1 < S) ? s + 1 : s;
        const size_t basen = (size_t)(b * S + sn) * A + h * 64;
        const float kn0 = Kx[basen + e0], kn1 = Kx[basen + e1];
        const float qn0 = Q[basen + e0],  qn1 = Q[basen + e1];
        const float vn0 = V[basen + e0],  vn1 = V[basen + e1];
        const float etan = ETA[(size_t)(b * S + sn) * H + h];
        if (s + 2 < S) {                 // pull s+2 toward the WGP caches
            const size_t basep = (size_t)(b * S + s + 2) * A + h * 64;
            __builtin_prefetch(&Kx[basep + e0], 0, 1);
            __builtin_prefetch(&Q[basep + e0], 0, 1);
            __builtin_prefetch(&V[basep + e0], 0, 1);
        }

        // Both mat-vecs + q.k before any cross-lane dependency.
        float z0 = bc0, z1 = bc1, zq0a = bc0, zq1a = bc1, qk = 0.f;
#pragma unroll
        for (int j = 0; j < 16; ++j) {   // ds_load_b128 broadcasts
            const float4 kk = skv4[j];
            const float4 qq = sqv4[j];
            kreg[4*j+0] = kk.x; kreg[4*j+1] = kk.y;
            kreg[4*j+2] = kk.z; kreg[4*j+3] = kk.w;
            z0   += kk.x*Wa[4*j] + kk.y*Wa[4*j+1] + kk.z*Wa[4*j+2] + kk.w*Wa[4*j+3];
            z1   += kk.x*Wb[4*j] + kk.y*Wb[4*j+1] + kk.z*Wb[4*j+2] + kk.w*Wb[4*j+3];
            zq0a += qq.x*Wa[4*j] + qq.y*Wa[4*j+1] + qq.z*Wa[4*j+2] + qq.w*Wa[4*j+3];
            zq1a += qq.x*Wb[4*j] + qq.y*Wb[4*j+1] + qq.z*Wb[4*j+2] + qq.w*Wb[4*j+3];
            qk   += kk.x*qq.x + kk.y*qq.y + kk.z*qq.z + kk.w*qq.w;
        }

        // LayerNorm stats of Z over the 64 columns (pure shfl reduction).
        float s1 = z0 + z1, s2 = z0*z0 + z1*z1;
        wave_red2(s1, s2);
        const float mu   = s1 * (1.0f / 64.0f);
        const float var  = s2 * (1.0f / 64.0f) - mu * mu;
        const float istd = rsqrtf(var + 1e-6f);
        const float zh0  = (z0 - mu) * istd, zh1 = (z1 - mu) * istd;

        // L2 grad through LN: target = v - k
        const float go0 = zh0 * gam0 + bet0 - (vc0 - kc0);
        const float go1 = zh1 * gam1 + bet1 - (vc1 - kc1);
        const float gx0 = go0 * gam0, gx1 = go1 * gam1;
        float g1 = gx0 + gx1, g2 = gx0 * zh0 + gx1 * zh1;
        wave_red2(g1, g2);
        const float gr0 = (64.0f * gx0 - g1 - zh0 * g2) * (istd * (1.0f / 64.0f));
        const float gr1 = (64.0f * gx1 - g1 - zh1 * g2) * (istd * (1.0f / 64.0f));

        // Rank-1 fast-weight update + refactored Zq.
        const float eg0 = etac * gr0, eg1 = etac * gr1;
        bc0 -= eg0; bc1 -= eg1;
#pragma unroll
        for (int d = 0; d < 64; ++d) {
            Wa[d] -= eg0 * kreg[d];
            Wb[d] -= eg1 * kreg[d];
        }
        const float zq0 = zq0a - eg0 * (qk + 1.0f);
        const float zq1 = zq1a - eg1 * (qk + 1.0f);

        // out = q + LN(Zq)
        float t1 = zq0 + zq1, t2 = zq0*zq0 + zq1*zq1;
        wave_red2(t1, t2);
        const float muq  = t1 * (1.0f / 64.0f);
        const float varq = t2 * (1.0f / 64.0f) - muq * muq;
        const float istq = rsqrtf(varq + 1e-6f);
        YS[baseo + e0] = qc0 + (zq0 - muq) * istq * gam0 + bet0;
        YS[baseo + e1] = qc1 + (zq1 - muq) * istq * gam1 + bet1;
        __syncthreads();                 // protect sk/sq before next staging

        kc0 = kn0; kc1 = kn1; qc0 = qn0; qc1 = qn1;
        vc0 = vn0; vc1 = vn1; etac = etan;
    }
}

// ---------------------------------------------------------------------------
// Host launcher
// ---------------------------------------------------------------------------
extern "C" void kernel_launch(void* const* d_in, const int* in_sizes, int n_in,
                              void* d_out, int out_size, void* d_ws, size_t ws_size,
                              hipStream_t stream)
{
    const float* x      = (const float*)d_in[0];
    const float* W_down = (const float*)d_in[1];
    const float* b_down = (const float*)d_in[2];
    const float* Wq     = (const float*)d_in[3];
    const float* Wk     = (const float*)d_in[4];
    const float* Wv     = (const float*)d_in[5];
    const float* Wo     = (const float*)d_in[6];
    const float* W1     = (const float*)d_in[7];
    const float* b1     = (const float*)d_in[8];
    const float* ttt_g  = (const float*)d_in[9];
    const float* ttt_b  = (const float*)d_in[10];
    const float* lr_w   = (const float*)d_in[11];
    const float* lr_b   = (const float*)d_in[12];
    const float* post_g = (const float*)d_in[13];
    const float* post_b = (const float*)d_in[14];
    const float* norm_g = (const float*)d_in[15];
    const float* norm_b = (const float*)d_in[16];
    const float* W_up   = (const float*)d_in[17];
    const float* b_up   = (const float*)d_in[18];
    float* out = (float*)d_out;

    const int B = 4, S = 1024, D = 768, A = 192, H = 3;
    const int M = B * S;                       // 4096 rows
    const size_t big = (size_t)M * A;          // 786432 floats

    float* ws    = (float*)d_ws;
    float* hbuf  = ws;                 // h, later reused for post-LN(u)
    float* qbuf  = hbuf + big;         // q, later reused for t2
    float* kbuf  = qbuf + big;         // k, later reused for final LN
    float* vbuf  = kbuf + big;         // v
    float* ysbuf = vbuf + big;         // scan output
    float* eta   = ysbuf + big;        // M*H

    const dim3 blk(128);

    // 1) h = silu(x @ W_down + b_down)
    wmma_gemm_f32<2><<<dim3(A / 16, M / 64), blk, 0, stream>>>(
        x, W_down, b_down, hbuf, M, A, D);

    // 2) q, k, v projections
    wmma_gemm_f32<0><<<dim3(A / 16, M / 64), blk, 0, stream>>>(
        hbuf, Wq, nullptr, qbuf, M, A, A);
    wmma_gemm_f32<0><<<dim3(A / 16, M / 64), blk, 0, stream>>>(
        hbuf, Wk, nullptr, kbuf, M, A, A);
    wmma_gemm_f32<0><<<dim3(A / 16, M / 64), blk, 0, stream>>>(
        hbuf, Wv, nullptr, vbuf, M, A, A);

    // 3) eta = sigmoid(h @ lr_w + lr_b) / hd
    eta_kernel<<<(M * H + 255) / 256, 256, 0, stream>>>(
        hbuf, lr_w, lr_b, eta, M, A, H);

    // 4) sequential TTT scan: one wave per (b,h) chain
    ttt_scan<<<B * H, 32, 0, stream>>>(
        qbuf, kbuf, vbuf, eta, W1, b1, ttt_g, ttt_b, ysbuf, S, H, A);

    // 5) u = LN(ys, post)   (reuse hbuf)
    ln_rows<<<M, A, 0, stream>>>(ysbuf, post_g, post_b, hbuf, A);

    // 6) t2 = u @ Wo        (reuse qbuf)
    wmma_gemm_f32<0><<<dim3(A / 16, M / 64), blk, 0, stream>>>(
        hbuf, Wo, nullptr, qbuf, M, A, A);

    // 7) w = LN(t2, norm)   (reuse kbuf)
    ln_rows<<<M, A, 0, stream>>>(qbuf, norm_g, norm_b, kbuf, A);

    // 8) out = w @ W_up + b_up
    wmma_gemm_f32<1><<<dim3(D / 16, M / 64), blk, 0, stream>>>(
        kbuf, W_up, b_up, out, M, D, A);
}